// ChamferDistance_14620068675781
// MI455X (gfx1250) — compile-verified
//
#include <hip/hip_runtime.h>

typedef __attribute__((ext_vector_type(2))) float v2f;
typedef __attribute__((ext_vector_type(8))) float v8f;

#define N_PTS 16384
#define CHUNKS 8
#define COLS_PER_CHUNK (N_PTS / CHUNKS)       // 2048
#define TILES_PER_CHUNK (COLS_PER_CHUNK / 16) // 128
#define STRIPS (N_PTS / 16)                   // 1024 row-strips per direction
#define WAVES_PER_BLOCK 8                     // 256 threads, wave32

// Seed outputs with +inf bit pattern every call (we merge with atomic-min and
// the harness does not re-poison between replays).
__global__ void chamfer_init(unsigned int* out) {
    int i = blockIdx.x * blockDim.x + threadIdx.x;
    if (i < 2 * N_PTS) out[i] = 0x7F800000u; // +inf as float bits
}

// Pack both clouds into lane-half-specific, WMMA-B-fragment-ready float4s:
//   lo[j] = (y0, y1, |y|^2, 0)   -> lanes 0..15  (B slots K=0,K=1)
//   hi[j] = (y2, 1.0, |y|^2, 0)  -> lanes 16..31 (B slots K=2,K=3)
// Layout in ws: [src_lo | src_hi | tgt_lo | tgt_hi], each N_PTS float4s. 1 MB.
__global__ void chamfer_pack(const float* __restrict__ src,
                             const float* __restrict__ tgt,
                             float4* __restrict__ pk) {
    int i = blockIdx.x * blockDim.x + threadIdx.x;
    if (i >= 2 * N_PTS) return;
    int cloud = (i < N_PTS) ? 0 : 1;
    int j = i - cloud * N_PTS;
    const float* p = (cloud ? tgt : src) + 3 * (size_t)j;
    float x = p[0], y = p[1], z = p[2];
    float s2 = x * x + y * y + z * z;
    float4* base = pk + (size_t)cloud * 2 * N_PTS;
    base[j]         = make_float4(x, y, s2, 0.0f); // lo half
    base[N_PTS + j] = make_float4(z, 1.0f, s2, 0.0f); // hi half
}

// One wave: 16-row strip of the query cloud vs a 2048-column chunk of the
// reference cloud. V_WMMA_F32_16X16X4_F32 with the embedding
//   A row = (-2x0, -2x1, -2x2, |x|^2),  B col = (y0, y1, y2, 1)
// gives |x|^2 - 2 x.y per 16x16 tile; |y|^2 (in q.z of the packed fragment)
// is folded into the running per-row min. B fragments come straight out of
// the load: the first two dwords of the lane-half-specific packed entry ARE
// the B operand registers — no per-tile selects.
__global__ __launch_bounds__(256) void chamfer_wmma_packed(
        const float* __restrict__ src, const float* __restrict__ tgt,
        const float4* __restrict__ pk, unsigned int* __restrict__ out) {
    const int wave = blockIdx.x * WAVES_PER_BLOCK + (threadIdx.x >> 5);
    const int lane = threadIdx.x & 31;

    const int dir   = wave / (STRIPS * CHUNKS);
    const int rem   = wave % (STRIPS * CHUNKS);
    const int strip = rem / CHUNKS;
    const int chunk = rem % CHUNKS;

    const int r  = lane & 15; // row (A) / column (B) within tile
    const int hi = lane >> 4; // lane-half: K=0/1 slots vs K=2/3 slots

    // A fragment from raw input (once per wave)
    const float* A_raw = dir ? tgt : src;
    const float* p = A_raw + (size_t)(strip * 16 + r) * 3;
    const float px = p[0], py = p[1], pz = p[2];
    const float x2 = px * px + py * py + pz * pz;
    v2f afrag;
    afrag.x = hi ? (-2.0f * pz) : (-2.0f * px); // K=2 : K=0
    afrag.y = hi ? x2           : (-2.0f * py); // K=3 : K=1

    v8f run;
#pragma unroll
    for (int k = 0; k < 8; ++k) run[k] = __builtin_inff();

    // Reference cloud, lane-half-specific packed stream
    const float4* bb = pk + (size_t)(dir ? 0 : 2 * N_PTS) + (size_t)hi * N_PTS
                          + (size_t)chunk * COLS_PER_CHUNK + r;

#pragma unroll 4
    for (int t = 0; t < TILES_PER_CHUNK; ++t) {
        float4 q = bb[(size_t)t * 16]; // global_load_b128; q.xy == B operand
        v2f bfrag;
        bfrag.x = q.x;
        bfrag.y = q.y;
        v8f c = {};
        c = __builtin_amdgcn_wmma_f32_16x16x4_f32(
                false, afrag, false, bfrag, (short)0, c, false, false);
#pragma unroll
        for (int k = 0; k < 8; ++k) run[k] = fminf(run[k], c[k] + q.z);
    }

    // Cross-lane min within each 16-lane half (xor offsets < 16 stay in-half).
#pragma unroll
    for (int off = 1; off < 16; off <<= 1) {
#pragma unroll
        for (int k = 0; k < 8; ++k)
            run[k] = fminf(run[k], __shfl_xor(run[k], off, 32));
    }

    // Lane 0: rows strip*16+0..7; lane 16: rows strip*16+8..15. Merge chunks
    // with u32 atomic-min (results clamped non-negative; non-negative IEEE
    // floats are order-isomorphic to their u32 bits).
    if (r == 0) {
        const int rbase = strip * 16 + hi * 8;
        unsigned int* o = out + dir * N_PTS;
#pragma unroll
        for (int k = 0; k < 8; ++k) {
            float v = run[k] < 0.0f ? 0.0f : run[k];
            atomicMin(&o[rbase + k], __float_as_uint(v));
        }
    }
}

// Fallback (no workspace): identical math, raw b96 loads + in-loop |y|^2.
__global__ __launch_bounds__(256) void chamfer_wmma_raw(
        const float* __restrict__ src, const float* __restrict__ tgt,
        unsigned int* __restrict__ out) {
    const int wave = blockIdx.x * WAVES_PER_BLOCK + (threadIdx.x >> 5);
    const int lane = threadIdx.x & 31;

    const int dir   = wave / (STRIPS * CHUNKS);
    const int rem   = wave % (STRIPS * CHUNKS);
    const int strip = rem / CHUNKS;
    const int chunk = rem % CHUNKS;

    const float* A_pts = dir ? tgt : src;
    const float* B_pts = dir ? src : tgt;
    unsigned int* o = out + dir * N_PTS;

    const int r  = lane & 15;
    const int hi = lane >> 4;

    const float* p = A_pts + (size_t)(strip * 16 + r) * 3;
    const float px = p[0], py = p[1], pz = p[2];
    const float x2 = px * px + py * py + pz * pz;
    v2f afrag;
    afrag.x = hi ? (-2.0f * pz) : (-2.0f * px);
    afrag.y = hi ? x2           : (-2.0f * py);

    v8f run;
#pragma unroll
    for (int k = 0; k < 8; ++k) run[k] = __builtin_inff();

    const float* bbase = B_pts + ((size_t)chunk * COLS_PER_CHUNK + (size_t)r) * 3;

#pragma unroll 2
    for (int t = 0; t < TILES_PER_CHUNK; ++t) {
        const float* q = bbase + (size_t)t * 16 * 3;
        const float qx = q[0], qy = q[1], qz = q[2];
        const float y2 = qx * qx + qy * qy + qz * qz;
        v2f bfrag;
        bfrag.x = hi ? qz   : qx;
        bfrag.y = hi ? 1.0f : qy;
        v8f c = {};
        c = __builtin_amdgcn_wmma_f32_16x16x4_f32(
                false, afrag, false, bfrag, (short)0, c, false, false);
#pragma unroll
        for (int k = 0; k < 8; ++k) run[k] = fminf(run[k], c[k] + y2);
    }

#pragma unroll
    for (int off = 1; off < 16; off <<= 1) {
#pragma unroll
        for (int k = 0; k < 8; ++k)
            run[k] = fminf(run[k], __shfl_xor(run[k], off, 32));
    }

    if (r == 0) {
        const int rbase = strip * 16 + hi * 8;
#pragma unroll
        for (int k = 0; k < 8; ++k) {
            float v = run[k] < 0.0f ? 0.0f : run[k];
            atomicMin(&o[rbase + k], __float_as_uint(v));
        }
    }
}

extern "C" void kernel_launch(void* const* d_in, const int* in_sizes, int n_in,
                              void* d_out, int out_size, void* d_ws, size_t ws_size,
                              hipStream_t stream) {
    const float* srcp = (const float*)d_in[0]; // [1, N, 3] source cloud
    const float* tgtp = (const float*)d_in[1]; // [1, N, 3] target cloud
    unsigned int* out = (unsigned int*)d_out;  // 2*N floats, fwd ++ bwd

    chamfer_init<<<(2 * N_PTS + 255) / 256, 256, 0, stream>>>(out);

    const int total_waves = 2 * STRIPS * CHUNKS;      // 16384
    const int blocks = total_waves / WAVES_PER_BLOCK; // 2048

    const size_t need = (size_t)4 * N_PTS * sizeof(float4); // 1 MB
    if (ws_size >= need) {
        float4* pk = (float4*)d_ws;
        chamfer_pack<<<(2 * N_PTS + 255) / 256, 256, 0, stream>>>(srcp, tgtp, pk);
        chamfer_wmma_packed<<<blocks, 256, 0, stream>>>(srcp, tgtp, pk, out);
    } else {
        chamfer_wmma_raw<<<blocks, 256, 0, stream>>>(srcp, tgtp, out);
    }
}